// MultiHeadContentAddressing_81003083203010
// MI455X (gfx1250) — compile-verified
//
#include <hip/hip_runtime.h>
#include <math.h>

#define B_   16
#define M_   32768
#define I_   512
#define H_   8
#define HD_  64
#define EPS_ 1e-8f
#define CH_  64                 // chunks per batch item
#define ROWS_ (M_ / CH_)        // 512 rows per chunk

typedef __attribute__((ext_vector_type(2))) float v2f;
typedef __attribute__((ext_vector_type(4))) float v4f;
typedef __attribute__((ext_vector_type(8))) float v8f;

// ---------------------------------------------------------------------------
// out[16 x 512] = A[16 x 512] @ W^T + bias, optional tanh.
// Uses V_WMMA_F32_16X16X4_F32 (full fp32), one wave per 16-column tile.
// A-frag: lane q=l%16 -> row q, K offset 2*(l/16). B[k][n] = W[n][k] so the
// B-frag load indexes W identically. D: row = v + 8*(l/16), col = q.
// ---------------------------------------------------------------------------
__global__ __launch_bounds__(32)
void gemm16_wmma(const float* __restrict__ A, const float* __restrict__ W,
                 const float* __restrict__ bias, float* __restrict__ out,
                 int do_tanh) {
  const int lane = threadIdx.x & 31;
  const int q    = lane & 15;
  const int half = lane >> 4;
  const int koff = 2 * half;
  const int n0   = blockIdx.x * 16;

  const float* arow = A + q * I_;
  const float* wrow = W + (n0 + q) * I_;

  v8f c = {};
  for (int k = 0; k < I_; k += 4) {
    v2f a = *(const v2f*)(arow + k + koff);
    v2f b = *(const v2f*)(wrow + k + koff);
    c = __builtin_amdgcn_wmma_f32_16x16x4_f32(
        /*neg_a=*/false, a, /*neg_b=*/false, b,
        /*c_mod=*/(short)0, c, /*reuse_a=*/false, /*reuse_b=*/false);
  }
  const float bv = bias[n0 + q];
#pragma unroll
  for (int v = 0; v < 8; ++v) {
    int   row = v + 8 * half;
    float val = c[v] + bv;
    if (do_tanh) val = tanhf(val);
    out[row * I_ + n0 + q] = val;
  }
}

// ---------------------------------------------------------------------------
// Per-head 64x64 projection + tanh + L2-normalize; beta folded in.
// kb[b,h,e] = beta[b,h] * tanh(pk[b,h,:] @ Wh[h,e,:] + bh[h,e]) / (||.||+eps)
// ---------------------------------------------------------------------------
__global__ __launch_bounds__(64)
void head_proj(const float* __restrict__ pk, const float* __restrict__ Wh,
               const float* __restrict__ bh, const float* __restrict__ beta,
               float* __restrict__ kb) {
  const int bhid = blockIdx.x;          // b*H + h
  const int h    = bhid % H_;
  const int b    = bhid / H_;
  const int e    = threadIdx.x;

  const float* x = pk + b * I_ + h * HD_;
  const float* w = Wh + (h * HD_ + e) * HD_;
  float acc = bh[h * HD_ + e];
  for (int d = 0; d < HD_; ++d) acc += x[d] * w[d];
  float hk = tanhf(acc);

  __shared__ float sh[64];
  sh[e] = hk * hk;
  __syncthreads();
  for (int s = 32; s > 0; s >>= 1) {
    if (e < s) sh[e] += sh[e + s];
    __syncthreads();
  }
  float nrm = sqrtf(sh[0]);
  kb[bhid * HD_ + e] = beta[bhid] * hk / (nrm + EPS_);
}

// ---------------------------------------------------------------------------
// Main 1-GiB scan. One workgroup per (b, chunk). 8 waves, each wave owns a
// row at a time (full 2 KB row -> all 8 heads), online softmax per head.
// Streaming loads are NON-TEMPORAL (TH=NT): memory is single-use, and NT
// keeps the 16 MB logit buffer + partials (re-read by later kernels) from
// being swept out of the 192 MB L2 by this 1 GiB stream.
// ---------------------------------------------------------------------------
__global__ __launch_bounds__(256)
void ca_scan(const float* __restrict__ mem, const float* __restrict__ kb,
             float* __restrict__ wout, float* __restrict__ part) {
  const int b     = blockIdx.x / CH_;
  const int chunk = blockIdx.x % CH_;
  const int wv    = threadIdx.x >> 5;
  const int lane  = threadIdx.x & 31;
  const int q     = lane & 15;
  const int half  = lane >> 4;

  // Preload this lane's slice of the (beta-scaled, normalized) key per head.
  float kreg[4][4];
#pragma unroll
  for (int it = 0; it < 4; ++it) {
    const int head = 2 * it + half;
    const float* kp = kb + (b * H_ + head) * HD_ + 4 * q;
    kreg[it][0] = kp[0]; kreg[it][1] = kp[1];
    kreg[it][2] = kp[2]; kreg[it][3] = kp[3];
  }

  float mx[4], ls[4], acc[4][4];
#pragma unroll
  for (int it = 0; it < 4; ++it) {
    mx[it] = -__builtin_inff();
    ls[it] = 0.f;
    acc[it][0] = acc[it][1] = acc[it][2] = acc[it][3] = 0.f;
  }

  const int r0 = chunk * ROWS_;
  for (int i = 0; i < ROWS_ / 8; ++i) {
    const int r = r0 + wv + 8 * i;
    const float* row = mem + (size_t)(b * M_ + r) * I_;
    __builtin_prefetch(row + 8 * I_ + 16 * lane, 0, 0);  // next row for this wave
#pragma unroll
    for (int it = 0; it < 4; ++it) {
      const int head = 2 * it + half;
      v4f v = __builtin_nontemporal_load(
                  (const v4f*)(row + 128 * it + 4 * lane));
      float dp = v.x * kreg[it][0] + v.y * kreg[it][1] +
                 v.z * kreg[it][2] + v.w * kreg[it][3];
      float nr = v.x * v.x + v.y * v.y + v.z * v.z + v.w * v.w;
#pragma unroll
      for (int m = 1; m <= 8; m <<= 1) {
        dp += __shfl_xor(dp, m, 32);
        nr += __shfl_xor(nr, m, 32);
      }
      float s = dp / (sqrtf(nr) + EPS_);          // = beta * cosine sim
      if (q == 0) wout[(size_t)(b * H_ + head) * M_ + r] = s;
      // online softmax update (state replicated across the 16-lane group)
      float nm   = fmaxf(mx[it], s);
      float corr = __expf(mx[it] - nm);
      float p    = __expf(s - nm);
      ls[it] = ls[it] * corr + p;
      acc[it][0] = acc[it][0] * corr + p * v.x;
      acc[it][1] = acc[it][1] * corr + p * v.y;
      acc[it][2] = acc[it][2] * corr + p * v.z;
      acc[it][3] = acc[it][3] * corr + p * v.w;
      mx[it] = nm;
    }
  }

  // Combine 8 waves within the workgroup via LDS.
  __shared__ float smax[8][8], sl[8][8], sacc[8][8][64];
#pragma unroll
  for (int it = 0; it < 4; ++it) {
    const int head = 2 * it + half;
    if (q == 0) { smax[wv][head] = mx[it]; sl[wv][head] = ls[it]; }
#pragma unroll
    for (int j = 0; j < 4; ++j) sacc[wv][head][4 * q + j] = acc[it][j];
  }
  __syncthreads();

  const int head = threadIdx.x >> 5;   // warp h handles head h
  const int l32  = threadIdx.x & 31;
  float gm = -__builtin_inff();
#pragma unroll
  for (int w = 0; w < 8; ++w) gm = fmaxf(gm, smax[w][head]);
  float L = 0.f, a0 = 0.f, a1 = 0.f;
#pragma unroll
  for (int w = 0; w < 8; ++w) {
    float e = __expf(smax[w][head] - gm);
    L  += sl[w][head] * e;
    a0 += sacc[w][head][l32] * e;
    a1 += sacc[w][head][l32 + 32] * e;
  }
  float* pp = part + (size_t)((b * H_ + head) * CH_ + chunk) * (HD_ + 2);
  if (l32 == 0) { pp[0] = gm; pp[1] = L; }
  pp[2 + l32]      = a0;
  pp[2 + l32 + 32] = a1;
}

// ---------------------------------------------------------------------------
// Combine chunk partials per (b,h): final max/L + reads[B,I].
// ---------------------------------------------------------------------------
__global__ __launch_bounds__(64)
void ca_reduce(const float* __restrict__ part, float* __restrict__ mgA,
               float* __restrict__ lA, float* __restrict__ reads) {
  const int bh = blockIdx.x;            // b*H + h
  const int t  = threadIdx.x;           // 0..63 (= head dim)
  const float* base = part + (size_t)bh * CH_ * (HD_ + 2);
  float gm = -__builtin_inff();
  for (int c = 0; c < CH_; ++c) gm = fmaxf(gm, base[c * (HD_ + 2)]);
  float L = 0.f, A = 0.f;
  for (int c = 0; c < CH_; ++c) {
    const float* p = base + c * (HD_ + 2);
    float e = __expf(p[0] - gm);
    L += p[1] * e;
    A += p[2 + t] * e;
  }
  if (t == 0) { mgA[bh] = gm; lA[bh] = L; }
  const int b = bh / H_, h = bh % H_;
  reads[b * I_ + h * HD_ + t] = A / L;  // [B, I] layout for the output GEMM
}

// ---------------------------------------------------------------------------
// Finalize softmax weights in place: w = exp(s - max) / L  (vectorized x4).
// These logits were written RT, so they should still be L2-resident.
// ---------------------------------------------------------------------------
__global__ __launch_bounds__(256)
void w_norm(float* __restrict__ w, const float* __restrict__ mgA,
            const float* __restrict__ lA) {
  const size_t idx = ((size_t)blockIdx.x * 256 + threadIdx.x) * 4;
  const int bh = (int)(idx >> 15);      // / M_ (M_ = 2^15)
  const float gm = mgA[bh];
  const float il = 1.f / lA[bh];
  v4f v = *(const v4f*)(w + idx);
  v.x = __expf(v.x - gm) * il;
  v.y = __expf(v.y - gm) * il;
  v.z = __expf(v.z - gm) * il;
  v.w = __expf(v.w - gm) * il;
  *(v4f*)(w + idx) = v;
}

// ---------------------------------------------------------------------------
extern "C" void kernel_launch(void* const* d_in, const int* in_sizes, int n_in,
                              void* d_out, int out_size, void* d_ws, size_t ws_size,
                              hipStream_t stream) {
  (void)in_sizes; (void)n_in; (void)out_size; (void)ws_size;
  const float* memory = (const float*)d_in[0];
  const float* key    = (const float*)d_in[1];
  const float* beta   = (const float*)d_in[2];
  const float* Wk     = (const float*)d_in[3];
  const float* bk     = (const float*)d_in[4];
  const float* Wh     = (const float*)d_in[5];
  const float* bhv    = (const float*)d_in[6];
  const float* Wo     = (const float*)d_in[7];
  const float* bo     = (const float*)d_in[8];

  float* out  = (float*)d_out;          // [B, I]
  float* wout = out + B_ * I_;          // [B, H, M]

  float* ws    = (float*)d_ws;
  float* kb    = ws;                                    // B*H*HD   = 8192
  float* pk    = kb + B_ * I_;                          // B*I      = 8192
  float* part  = pk + B_ * I_;                          // B*H*CH*66
  float* mgA   = part + (size_t)B_ * H_ * CH_ * (HD_ + 2);
  float* lA    = mgA + B_ * H_;
  float* reads = lA + B_ * H_;                          // B*I

  gemm16_wmma<<<I_ / 16, 32, 0, stream>>>(key, Wk, bk, pk, /*tanh=*/1);
  head_proj  <<<B_ * H_, 64, 0, stream>>>(pk, Wh, bhv, beta, kb);
  ca_scan    <<<B_ * CH_, 256, 0, stream>>>(memory, kb, wout, part);
  ca_reduce  <<<B_ * H_, 64, 0, stream>>>(part, mgA, lA, reads);
  w_norm     <<<(B_ * H_ * M_) / (256 * 4), 256, 0, stream>>>(wout, mgA, lA);
  gemm16_wmma<<<I_ / 16, 32, 0, stream>>>(reads, Wo, bo, out, /*tanh=*/0);
}